// SmoothLossV2_652835029198
// MI455X (gfx1250) — compile-verified
//
#include <hip/hip_runtime.h>
#include <hip/hip_bf16.h>

#define NSEG 8
#define SEG 4096
#define NPTS (NSEG * SEG)
#define KNN 3
#define TILE 16
#define BLOCK 256
#define WAVES (BLOCK / 32)
#define ROWS_PER_BLOCK (WAVES * TILE)          /* 128 */
#define BLOCKS_PER_SEG (SEG / ROWS_PER_BLOCK)  /* 32  */
#define CTILES (SEG / TILE)                    /* 256 */
#define BIGF 3.4e38f

typedef float v2f __attribute__((ext_vector_type(2)));
typedef float v8f __attribute__((ext_vector_type(8)));
typedef unsigned long long u64;

__device__ __forceinline__ u64 umin64(u64 a, u64 b) { return a < b ? a : b; }
__device__ __forceinline__ u64 umax64(u64 a, u64 b) { return a < b ? b : a; }

// Branch-free insert of candidate c into sorted triple k0<=k1<=k2:
// 5 u64 min/max ops -> v_cmp_lt_u64 + v_cndmask pairs, no control flow.
__device__ __forceinline__ void top3_insert_u64(u64 c, u64& k0, u64& k1, u64& k2) {
  u64 h0 = umax64(k0, c);
  k0 = umin64(k0, c);
  u64 h1 = umax64(k1, h0);
  k1 = umin64(k1, h0);
  k2 = umin64(k2, h1);
}

__global__ void zero_acc_kernel(float* acc) {
  if (threadIdx.x == 0 && blockIdx.x == 0) acc[0] = 0.0f;
}

__global__ __launch_bounds__(256) void normalize_kernel(
    const float* __restrict__ pred, float* __restrict__ pn, int n) {
  int i = blockIdx.x * blockDim.x + threadIdx.x;
  if (i >= n) return;
  float x = pred[3 * i + 0], y = pred[3 * i + 1], z = pred[3 * i + 2];
  float inp = sqrtf(x * x + y * y + z * z + 1e-8f);
  float s = 1.0f / (inp + 1e-10f);
  pn[3 * i + 0] = x * s;
  pn[3 * i + 1] = y * s;
  pn[3 * i + 2] = z * s;
}

__global__ __launch_bounds__(BLOCK) void knn_loss_kernel(
    const float* __restrict__ coord, const float* __restrict__ pn,
    const float* __restrict__ tgt, float* __restrict__ acc) {
  __shared__ float s_cx[SEG];
  __shared__ float s_cy[SEG];
  __shared__ float s_cz[SEG];
  __shared__ int s_nbr[ROWS_PER_BLOCK][KNN];
  __shared__ float s_red[BLOCK];

  const int tid = threadIdx.x;
  const int lane = tid & 31;
  const int wave = tid >> 5;
  const int seg = blockIdx.x / BLOCKS_PER_SEG;
  const int rowBase = (blockIdx.x % BLOCKS_PER_SEG) * ROWS_PER_BLOCK;
  const float* cseg = coord + (size_t)seg * SEG * 3;

  // Stage this segment's coordinates into LDS (48 KB).
  for (int t = tid; t < SEG; t += BLOCK) {
    s_cx[t] = cseg[3 * t + 0];
    s_cy[t] = cseg[3 * t + 1];
    s_cz[t] = cseg[3 * t + 2];
  }
  __syncthreads();

  const bool hi = (lane >= 16);
  const int nlo = lane & 15;
  const int r0 = rowBase + wave * TILE;  // segment-local first row of this wave's tile

  // A-matrix (16x4 f32): lanes 0-15 carry K=0,1 (x,y) of row r0+nlo;
  // lanes 16-31 carry K=2,3 (z,0) of the same rows.
  v2f A;
  {
    int ar = r0 + nlo;
    float ax = s_cx[ar], ay = s_cy[ar], az = s_cz[ar];
    A.x = hi ? az : ax;
    A.y = hi ? 0.0f : ay;
  }

  // |c_i|^2 for the 8 D-slot rows this lane observes (M = v + hi*8).
  float sqi[8];
#pragma unroll
  for (int v = 0; v < 8; ++v) {
    int r = r0 + v + (hi ? 8 : 0);
    float x = s_cx[r], y = s_cy[r], z = s_cz[r];
    sqi[v] = x * x + y * y + z * z;
  }

  // Top-3 as packed (d2_bits << 32 | col) keys, sorted ascending.
  // d2 >= 0 so float bits are order-monotonic; low-word col reproduces
  // top_k's lower-index tie-break.
  u64 kk[8][3];
#pragma unroll
  for (int v = 0; v < 8; ++v) {
    kk[v][0] = ~0ull; kk[v][1] = ~0ull; kk[v][2] = ~0ull;
  }

  for (int ct = 0; ct < CTILES; ++ct) {
    const int jc = ct * TILE + nlo;  // segment-local column this lane owns
    float bx = s_cx[jc], by = s_cy[jc], bz = s_cz[jc];
    v2f B;
    B.x = hi ? bz : bx;
    B.y = hi ? 0.0f : by;
    float sqj = bx * bx + by * by + bz * bz;
    v8f C = {};
    v8f D = __builtin_amdgcn_wmma_f32_16x16x4_f32(
        /*neg_a=*/false, A, /*neg_b=*/false, B,
        /*c_mod=*/(short)0, C, /*reuse_a=*/false, /*reuse_b=*/false);
#pragma unroll
    for (int v = 0; v < 8; ++v) {
      float d2 = fmaf(-2.0f, D[v], sqi[v] + sqj);
      d2 = fmaxf(d2, 0.0f);
      const int ig = r0 + v + (hi ? 8 : 0);
      d2 = (jc == ig) ? BIGF : d2;  // exclude self, no divergence
      u64 key = ((u64)__float_as_uint(d2) << 32) | (unsigned)jc;
      top3_insert_u64(key, kk[v][0], kk[v][1], kk[v][2]);
    }
  }

  // Butterfly merge of sorted triples across the 16 lanes of each half-wave.
#pragma unroll
  for (int m = 1; m <= 8; m <<= 1) {
#pragma unroll
    for (int v = 0; v < 8; ++v) {
      u64 p0 = __shfl_xor(kk[v][0], m, 32);
      u64 p1 = __shfl_xor(kk[v][1], m, 32);
      u64 p2 = __shfl_xor(kk[v][2], m, 32);
      top3_insert_u64(p0, kk[v][0], kk[v][1], kk[v][2]);
      top3_insert_u64(p1, kk[v][0], kk[v][1], kk[v][2]);
      top3_insert_u64(p2, kk[v][0], kk[v][1], kk[v][2]);
    }
  }

  // One lane per half publishes the 8 rows it holds.
  if (nlo == 0) {
    int base = wave * TILE + (hi ? 8 : 0);
#pragma unroll
    for (int v = 0; v < 8; ++v) {
      s_nbr[base + v][0] = (int)(unsigned)kk[v][0];
      s_nbr[base + v][1] = (int)(unsigned)kk[v][1];
      s_nbr[base + v][2] = (int)(unsigned)kk[v][2];
    }
  }
  __syncthreads();

  // Similarity phase: one thread per row of this block.
  float lsum = 0.0f;
  if (tid < ROWS_PER_BLOCK) {
    int ig = seg * SEG + rowBase + tid;
    float px = pn[3 * ig + 0], py = pn[3 * ig + 1], pz = pn[3 * ig + 2];
    float tx = tgt[3 * ig + 0], ty = tgt[3 * ig + 1], tz = tgt[3 * ig + 2];
#pragma unroll
    for (int k = 0; k < KNN; ++k) {
      int j = seg * SEG + s_nbr[tid][k];
      float sim = fabsf(px * pn[3 * j + 0] + py * pn[3 * j + 1] + pz * pn[3 * j + 2]);
      sim = fminf(sim, 1.0f);
      float ts = fabsf(tx * tgt[3 * j + 0] + ty * tgt[3 * j + 1] + tz * tgt[3 * j + 2]);
      ts = fminf(ts, 1.0f);
      float df = sim - ts;
      lsum += df * df;
    }
  }
  s_red[tid] = lsum;
  __syncthreads();
  for (int s = BLOCK / 2; s > 0; s >>= 1) {
    if (tid < s) s_red[tid] += s_red[tid + s];
    __syncthreads();
  }
  if (tid == 0) atomicAdd(acc, s_red[0]);
}

__global__ void finalize_kernel(const float* __restrict__ acc,
                                float* __restrict__ out) {
  if (threadIdx.x == 0 && blockIdx.x == 0)
    out[0] = acc[0] * (1.0f / (float)(NSEG * SEG * KNN));
}

extern "C" void kernel_launch(void* const* d_in, const int* in_sizes, int n_in,
                              void* d_out, int out_size, void* d_ws, size_t ws_size,
                              hipStream_t stream) {
  (void)in_sizes; (void)n_in; (void)out_size; (void)ws_size;
  const float* pred = (const float*)d_in[0];
  const float* coord = (const float*)d_in[1];
  const float* target = (const float*)d_in[2];
  // d_in[3] ("nums") only encodes NSEG; unused.

  float* ws = (float*)d_ws;
  float* acc = ws;        // 1 float accumulator
  float* pn = ws + 16;    // 64B-aligned; NPTS*3 floats of pred_norm

  zero_acc_kernel<<<1, 1, 0, stream>>>(acc);
  normalize_kernel<<<(NPTS + 255) / 256, 256, 0, stream>>>(pred, pn, NPTS);
  knn_loss_kernel<<<NSEG * BLOCKS_PER_SEG, BLOCK, 0, stream>>>(coord, pn, target, acc);
  finalize_kernel<<<1, 1, 0, stream>>>(acc, (float*)d_out);
}